// vector_18305150615824
// MI455X (gfx1250) — compile-verified
//
#include <hip/hip_runtime.h>
#include <math.h>

typedef __attribute__((ext_vector_type(2))) float v2f;
typedef __attribute__((ext_vector_type(8))) float v8f;

#define N_BASIS 10
#define HID 20
#define WAVES_PER_BLOCK 8
#define BLOCK (WAVES_PER_BLOCK * 32)

// D(16x16,f32) = A(16x4,f32) x B(4x16,f32) + C   -- exact fp32
static __device__ __forceinline__ v8f wmma_f32(v2f a, v2f b, v8f c) {
  return __builtin_amdgcn_wmma_f32_16x16x4_f32(
      /*neg_a=*/false, a, /*neg_b=*/false, b,
      /*c_mod=*/(short)0, c, /*reuse_a=*/false, /*reuse_b=*/false);
}

// single-instruction ReLU (inputs are never sNaN; skip LLVM's canonicalize)
static __device__ __forceinline__ float relu1(float x) {
  float r;
  asm("v_max_num_f32 %0, 0, %1" : "=v"(r) : "v"(x));
  return r;
}

// v2f = (v[i], v[i+1]); indices become constants after unroll
static __device__ __forceinline__ v2f pairf(v8f v, int i) {
  v2f r; r[0] = v[i]; r[1] = v[i + 1]; return r;
}

__global__ __launch_bounds__(BLOCK) void moe_hat_kernel(
    const v2f* __restrict__ x, const float* __restrict__ tp,
    const float* __restrict__ W1, const float* __restrict__ b1,
    const float* __restrict__ W2, const float* __restrict__ b2,
    const float* __restrict__ W3, const float* __restrict__ b3,
    v2f* __restrict__ out, int npoints)
{
  const int lane = threadIdx.x & 31;
  const int n0   = lane & 15;   // M row (A operand) / N point column (B, C, D)
  const int hh   = lane >> 4;   // lane half: selects K sub-pair / M+8 row

  // ---- hat-basis phi(t): at most two live experts (others weighted 0.0) ----
  const float t = *tp;
  int t0 = (int)rintf(t * (float)N_BASIS);      // jnp.round = RNE = rintf
  t0 = t0 < 0 ? 0 : (t0 > N_BASIS ? N_BASIS : t0);
  const float t0f = (float)t0;
  const float bw0 = ((t0f + 1.0f) / (float)N_BASIS - t) * (float)N_BASIS;
  const float bw1 = (t0 < N_BASIS) ? (t - t0f / (float)N_BASIS) * (float)N_BASIS : 0.0f;
  const int   eidx[2] = { t0, (t0 + 1 < N_BASIS) ? t0 + 1 : N_BASIS };
  const float ew[2]   = { bw0, bw1 };

  // ---- hidden-unit permutation so D registers feed the next WMMA directly --
  // K-slot c = 4k+2h+r is stored at D-row: 2k+r+8h (k<4), or 16+r+8h (k==4).
  // Hence next-layer B operand for kstep k is exactly (d[2k], d[2k+1]) (k<4)
  // or (d_t1[0], d_t1[1]) (k==4).  Row m holds unit c(m):
  //   m in 0..15        : c = 4*((m&7)>>1) + 2*(m>>3) + (m&1)
  //   m in {16,17,24,25}: c = 16 + 2*(m>=24) + (m&1)
  const int  u_t0 = 4 * ((n0 & 7) >> 1) + 2 * (n0 >> 3) + (n0 & 1);
  const bool t1v  = (n0 == 0) | (n0 == 1) | (n0 == 8) | (n0 == 9);
  const int  u_t1 = 16 + 2 * (n0 >> 3) + (n0 & 1);

  // ---- wave-resident A operands (weights) + bias C registers --------------
  // A(16x4) layout: lane n0 = M row; element (r, half hh) = K = 4k + 2hh + r.
  v2f A1w[2][2];        // [expert][Mtile]  layer1: K=0,1 -> W1; K=2 -> bias
  v2f A2w[2][2][5];     // [expert][Mtile][kstep]
  v2f A3w[2][5];        // [expert][kstep]  (M rows 0,1 valid = outputs)
  v8f cb2t0[2], cb2t1[2], cb3[2];

  #pragma unroll
  for (int e = 0; e < 2; ++e) {
    const int ex = eidx[e];
    #pragma unroll
    for (int mt = 0; mt < 2; ++mt) {
      const int  u  = mt ? u_t1 : u_t0;
      const bool uv = mt ? t1v  : true;
      #pragma unroll
      for (int r = 0; r < 2; ++r) {
        float w = 0.0f;
        if (uv) {
          if (hh == 0)     w = W1[ex * HID * 2 + u * 2 + r];  // K = 0,1
          else if (r == 0) w = b1[ex * HID + u];              // K = 2 (bias)
        }
        A1w[e][mt][r] = w;
      }
      #pragma unroll
      for (int k = 0; k < 5; ++k)
        #pragma unroll
        for (int r = 0; r < 2; ++r) {
          const int kk = 4 * k + 2 * hh + r;                  // < 20 always
          A2w[e][mt][k][r] = uv ? W2[ex * HID * HID + u * HID + kk] : 0.0f;
        }
    }
    #pragma unroll
    for (int k = 0; k < 5; ++k)
      #pragma unroll
      for (int r = 0; r < 2; ++r) {
        const int kk = 4 * k + 2 * hh + r;
        A3w[e][k][r] = (n0 < 2) ? W3[ex * 2 * HID + n0 * HID + kk] : 0.0f;
      }
    #pragma unroll
    for (int v = 0; v < 8; ++v) {
      cb2t0[e][v] = b2[ex * HID + 4 * (v >> 1) + 2 * hh + (v & 1)];
      cb2t1[e][v] = (v < 2) ? b2[ex * HID + 16 + 2 * hh + v] : 0.0f;
      cb3[e][v]   = (hh == 0 && v < 2) ? b3[ex * 2 + v] : 0.0f;
    }
  }

  v8f czero;
  #pragma unroll
  for (int v = 0; v < 8; ++v) czero[v] = 0.0f;

  const int ntiles = npoints >> 4;
  const int gw = blockIdx.x * WAVES_PER_BLOCK + (threadIdx.x >> 5);
  const int nw = gridDim.x * WAVES_PER_BLOCK;

  // B1 = X^T (2x16) + bias feed: half0 lanes carry (x0,x1) at K=0,1;
  // half1 lanes carry (1,0) at K=2,3 so A's K=2 bias column passes through.
  // Double-buffered: the next tile's x is loaded while this tile computes.
  v2f bx_next = { 1.0f, 0.0f };
  if (gw < ntiles && hh == 0) bx_next = x[(gw << 4) + n0];

  for (int tile = gw; tile < ntiles; tile += nw) {
    const v2f bx = bx_next;
    const int nt = tile + nw;
    v2f pf = { 1.0f, 0.0f };
    if (hh == 0 && nt < ntiles) pf = x[(nt << 4) + n0];   // prefetch next
    bx_next = pf;

    float ya = 0.0f, yb = 0.0f;

    #pragma unroll
    for (int e = 0; e < 2; ++e) {
      // ---- layer 1: H1^T = relu(W1 @ X^T + b1), rows permuted ----
      v8f h0 = wmma_f32(A1w[e][0], bx, czero);
      v8f h1 = wmma_f32(A1w[e][1], bx, czero);
      #pragma unroll
      for (int v = 0; v < 8; ++v) h0[v] = relu1(h0[v]);
      h1[0] = relu1(h1[0]);
      h1[1] = relu1(h1[1]);

      // ---- layer 2: H2^T = relu(W2 @ H1^T + b2), rows permuted ----
      v8f d0 = cb2t0[e];
      v8f d1 = cb2t1[e];
      #pragma unroll
      for (int k = 0; k < 4; ++k) {
        v2f bk = pairf(h0, 2 * k);
        d0 = wmma_f32(A2w[e][0][k], bk, d0);
        d1 = wmma_f32(A2w[e][1][k], bk, d1);
      }
      {
        v2f bk = pairf(h1, 0);
        d0 = wmma_f32(A2w[e][0][4], bk, d0);
        d1 = wmma_f32(A2w[e][1][4], bk, d1);
      }
      #pragma unroll
      for (int v = 0; v < 8; ++v) d0[v] = relu1(d0[v]);
      d1[0] = relu1(d1[0]);
      d1[1] = relu1(d1[1]);

      // ---- layer 3: Y^T = W3 @ H2^T + b3 (M rows 0,1 valid) ----
      v8f y = cb3[e];
      #pragma unroll
      for (int k = 0; k < 4; ++k)
        y = wmma_f32(A3w[e][k], pairf(d0, 2 * k), y);
      y = wmma_f32(A3w[e][4], pairf(d1, 0), y);

      ya = fmaf(ew[e], y[0], ya);
      yb = fmaf(ew[e], y[1], yb);
    }

    // D3 layout: half0 lane n0 holds (y0,y1) of point n0 -> coalesced 8B store
    if (hh == 0) {
      v2f yo = { ya, yb };
      out[(tile << 4) + n0] = yo;
    }
  }
}

extern "C" void kernel_launch(void* const* d_in, const int* in_sizes, int n_in,
                              void* d_out, int out_size, void* d_ws, size_t ws_size,
                              hipStream_t stream) {
  const v2f*   x  = (const v2f*)  d_in[0];
  const float* t  = (const float*)d_in[1];
  const float* W1 = (const float*)d_in[2];
  const float* b1 = (const float*)d_in[3];
  const float* W2 = (const float*)d_in[4];
  const float* b2 = (const float*)d_in[5];
  const float* W3 = (const float*)d_in[6];
  const float* b3 = (const float*)d_in[7];
  v2f* out = (v2f*)d_out;

  const int npoints = in_sizes[0] / 2;
  const int ntiles  = npoints / 16;
  // ~16 tiles per wave amortizes the per-wave weight-operand build
  int waves  = (ntiles + 15) / 16;
  int blocks = (waves + WAVES_PER_BLOCK - 1) / WAVES_PER_BLOCK;
  if (blocks < 1) blocks = 1;
  if (blocks > 65535) blocks = 65535;

  moe_hat_kernel<<<blocks, BLOCK, 0, stream>>>(x, t, W1, b1, W2, b2, W3, b3,
                                               out, npoints);
}